// GCN_67413806678429
// MI455X (gfx1250) — compile-verified
//
#include <hip/hip_runtime.h>

#define HID 64
#define WPAD 4   // LDS row padding: bank = (4*col + k) % 64 -> conflict-free b64 reads

typedef __attribute__((ext_vector_type(2))) float v2f;
typedef __attribute__((ext_vector_type(8))) float v8f;

// Hardware fp32 atomic add (global_atomic_add_f32); agg buffers are L2-resident.
__device__ __forceinline__ void atomicAddF32(float* p, float v) {
    unsafeAtomicAdd(p, v);
}

// ---------------- Layer 1: scalar edge scatter (IN_DIM == 1) ----------------
__global__ __launch_bounds__(256) void l1_edge(const int* __restrict__ src,
                                               const int* __restrict__ dst,
                                               const float* __restrict__ ew,
                                               const float* __restrict__ x,
                                               float* __restrict__ s1, int E) {
    int e = blockIdx.x * 256 + threadIdx.x;
    if (e >= E) return;
    atomicAddF32(&s1[dst[e]], x[src[e]] * ew[e]);
}

// Layer 1 node update: rank-1 outer products + bias + relu -> h1[N,64]
__global__ __launch_bounds__(256) void l1_node(const float* __restrict__ x,
                                               const float* __restrict__ s1,
                                               const float* __restrict__ W1rel,
                                               const float* __restrict__ b1,
                                               const float* __restrict__ W1root,
                                               float* __restrict__ h1, int n) {
    int idx = blockIdx.x * 256 + threadIdx.x;
    if (idx >= n * HID) return;
    int i = idx >> 6, f = idx & 63;
    float v = s1[i] * W1rel[f] + b1[f] + x[i] * W1root[f];
    h1[idx] = v > 0.f ? v : 0.f;
}

// ---------------- Layer 2: 64-wide edge gather/scatter-add ----------------
// 16 threads per edge, float4 per thread: coalesced 256B row gather + 4 atomics.
__global__ __launch_bounds__(256) void l2_edge(const int* __restrict__ src,
                                               const int* __restrict__ dst,
                                               const float* __restrict__ ew,
                                               const float* __restrict__ h1,
                                               float* __restrict__ agg, int E) {
    int tid = blockIdx.x * 256 + threadIdx.x;
    if (tid >= E * 16) return;
    int e = tid >> 4;
    int f = (tid & 15) << 2;
    float w = ew[e];
    int s = src[e], d = dst[e];
    const float4 hv = *(const float4*)(h1 + (size_t)s * HID + f);
    float* ap = agg + (size_t)d * HID + f;
    atomicAddF32(ap + 0, hv.x * w);
    atomicAddF32(ap + 1, hv.y * w);
    atomicAddF32(ap + 2, hv.z * w);
    atomicAddF32(ap + 3, hv.w * w);
}

// ---------------- Layer 2 node: WMMA GEMM, fp32, fused bias+relu ----------------
// h2 = relu([agg2 | h1] @ [W2_rel ; W2_root] + b2)   (M=N_nodes, N=64, K=128)
// One wave per 16-row tile; 4 x v8f accumulators cover N=64; K stepped by 4
// via V_WMMA_F32_16X16X4_F32. Weights stored TRANSPOSED in LDS so each B
// fragment (W[kk][n], W[kk+1][n]) is one contiguous 8B-aligned ds_load_b64.
__global__ __launch_bounds__(128) void l2_gemm(const float* __restrict__ agg,
                                               const float* __restrict__ h1,
                                               const float* __restrict__ Wrel,
                                               const float* __restrict__ Wroot,
                                               const float* __restrict__ b2,
                                               float* __restrict__ h2, int n) {
    __shared__ __align__(16) float sWt[2][HID][HID + WPAD];  // [half][col][row]
    __shared__ float sB[HID];

    int t = threadIdx.x;
    for (int idx = t; idx < HID * HID; idx += 128) {
        int r = idx >> 6, c = idx & 63;        // coalesced global read, LDS transpose
        sWt[0][c][r] = Wrel[idx];
        sWt[1][c][r] = Wroot[idx];
    }
    if (t < HID) sB[t] = b2[t];
    __syncthreads();

    int lane = t & 31;
    int wave = t >> 5;
    int m0 = (blockIdx.x * 4 + wave) << 4;    // 16-row tile per wave
    if (m0 >= n) return;                      // wave-uniform: EXEC stays all-1s

    int mrow = lane & 15;                     // A-frag row (ISA 16x4 f32 layout)
    int kh   = (lane >> 4) << 1;              // lanes 16-31 hold K+2,K+3
    int nn   = lane & 15;

    v8f acc0 = {0.f,0.f,0.f,0.f,0.f,0.f,0.f,0.f};
    v8f acc1 = {0.f,0.f,0.f,0.f,0.f,0.f,0.f,0.f};
    v8f acc2 = {0.f,0.f,0.f,0.f,0.f,0.f,0.f,0.f};
    v8f acc3 = {0.f,0.f,0.f,0.f,0.f,0.f,0.f,0.f};

    const float* Abase[2] = { agg + (size_t)(m0 + mrow) * HID,
                              h1  + (size_t)(m0 + mrow) * HID };
    for (int half = 0; half < 2; ++half) {
        const float* A = Abase[half];
        const float (*W)[HID + WPAD] = sWt[half];
        for (int k = 0; k < HID; k += 4) {
            int kk = k + kh;                              // even -> 8B aligned
            v2f a  = *(const v2f*)(A + kk);               // global_load_b64
            v2f b0 = *(const v2f*)&W[nn][kk];             // ds_load_b64, no repack
            v2f b1 = *(const v2f*)&W[16 + nn][kk];
            v2f b2v = *(const v2f*)&W[32 + nn][kk];
            v2f b3v = *(const v2f*)&W[48 + nn][kk];
            acc0 = __builtin_amdgcn_wmma_f32_16x16x4_f32(false, a, false, b0,  (short)0, acc0, false, false);
            acc1 = __builtin_amdgcn_wmma_f32_16x16x4_f32(false, a, false, b1,  (short)0, acc1, false, false);
            acc2 = __builtin_amdgcn_wmma_f32_16x16x4_f32(false, a, false, b2v, (short)0, acc2, false, false);
            acc3 = __builtin_amdgcn_wmma_f32_16x16x4_f32(false, a, false, b3v, (short)0, acc3, false, false);
        }
    }

    // C/D layout: VGPR v -> row v + 8*(lane/16), col = lane%16 (+16 per N tile)
    int rsel = (lane >> 4) << 3;
    #pragma unroll
    for (int v = 0; v < 8; ++v) {
        size_t row = (size_t)(m0 + v + rsel);
        float* outr = h2 + row * HID;
        float y0 = acc0[v] + sB[nn];
        float y1 = acc1[v] + sB[16 + nn];
        float y2 = acc2[v] + sB[32 + nn];
        float y3 = acc3[v] + sB[48 + nn];
        outr[nn]      = y0 > 0.f ? y0 : 0.f;
        outr[16 + nn] = y1 > 0.f ? y1 : 0.f;
        outr[32 + nn] = y2 > 0.f ? y2 : 0.f;
        outr[48 + nn] = y3 > 0.f ? y3 : 0.f;
    }
}

// ---------------- Layer 3 node: fold 64->1 projections before the scatter ----
// yrel[i] = h2[i].W3_rel ;  out[i] = b3 + h2[i].W3_root  (one wave per node)
__global__ __launch_bounds__(128) void l3_node(const float* __restrict__ h2,
                                               const float* __restrict__ W3rel,
                                               const float* __restrict__ W3root,
                                               const float* __restrict__ b3,
                                               float* __restrict__ out,
                                               float* __restrict__ yrel, int n) {
    int lane = threadIdx.x & 31;
    int wave = threadIdx.x >> 5;
    int i = blockIdx.x * 4 + wave;
    if (i >= n) return;
    const float2 hv = ((const float2*)(h2 + (size_t)i * HID))[lane];
    const float2 wr = ((const float2*)W3rel)[lane];
    const float2 wo = ((const float2*)W3root)[lane];
    float rel = hv.x * wr.x + hv.y * wr.y;
    float rt  = hv.x * wo.x + hv.y * wo.y;
    for (int off = 16; off > 0; off >>= 1) {
        rel += __shfl_xor(rel, off, 32);
        rt  += __shfl_xor(rt,  off, 32);
    }
    if (lane == 0) {
        yrel[i] = rel;
        out[i]  = rt + b3[0];
    }
}

// Layer 3 edge: scalar scatter of the projected values
__global__ __launch_bounds__(256) void l3_edge(const int* __restrict__ src,
                                               const int* __restrict__ dst,
                                               const float* __restrict__ ew,
                                               const float* __restrict__ yrel,
                                               float* __restrict__ out, int E) {
    int e = blockIdx.x * 256 + threadIdx.x;
    if (e >= E) return;
    atomicAddF32(&out[dst[e]], yrel[src[e]] * ew[e]);
}

extern "C" void kernel_launch(void* const* d_in, const int* in_sizes, int n_in,
                              void* d_out, int out_size, void* d_ws, size_t ws_size,
                              hipStream_t stream) {
    const float* x      = (const float*)d_in[0];
    const int*   ei     = (const int*)  d_in[1];
    const float* ew     = (const float*)d_in[2];
    const float* W1rel  = (const float*)d_in[3];
    const float* b1     = (const float*)d_in[4];
    const float* W1root = (const float*)d_in[5];
    const float* W2rel  = (const float*)d_in[6];
    const float* b2     = (const float*)d_in[7];
    const float* W2root = (const float*)d_in[8];
    const float* W3rel  = (const float*)d_in[9];
    const float* b3     = (const float*)d_in[10];
    const float* W3root = (const float*)d_in[11];
    float* out = (float*)d_out;

    int N = in_sizes[0];      // x is [N, 1]
    int E = in_sizes[2];      // edge_weight is [E]
    const int* src = ei;
    const int* dst = ei + E;

    // Workspace layout (floats): s1[N] | yrel[N] | h1[64N] | agg2[64N] | h2[64N]
    float* ws   = (float*)d_ws;
    float* s1   = ws;
    float* yrel = ws + (size_t)N;
    float* h1   = ws + 2 * (size_t)N;
    float* agg2 = h1 + (size_t)HID * N;
    float* h2   = agg2 + (size_t)HID * N;

    hipMemsetAsync(s1,   0, (size_t)N * sizeof(float), stream);
    hipMemsetAsync(agg2, 0, (size_t)N * HID * sizeof(float), stream);

    l1_edge<<<(E + 255) / 256, 256, 0, stream>>>(src, dst, ew, x, s1, E);
    l1_node<<<((size_t)N * HID + 255) / 256, 256, 0, stream>>>(x, s1, W1rel, b1, W1root, h1, N);
    l2_edge<<<((size_t)E * 16 + 255) / 256, 256, 0, stream>>>(src, dst, ew, h1, agg2, E);
    l2_gemm<<<(N + 63) / 64, 128, 0, stream>>>(agg2, h1, W2rel, W2root, b2, h2, N);
    l3_node<<<(N + 3) / 4, 128, 0, stream>>>(h2, W3rel, W3root, b3, out, yrel, N);
    l3_edge<<<(E + 255) / 256, 256, 0, stream>>>(src, dst, ew, yrel, out, E);
}